// VQC_Block_86388972191849
// MI455X (gfx1250) — compile-verified
//
#include <hip/hip_runtime.h>
#include <math.h>

typedef __attribute__((ext_vector_type(2))) float v2f;
typedef __attribute__((ext_vector_type(8))) float v8f;

#define NQ      12
#define NSTATE  4096      // 2^12
#define NLAYERS 4
#define NTHREADS 256      // 8 waves of 32
#define NGATES  (NQ + NLAYERS * NQ)   // 12 RX + 48 Rot = 60

// Build 16x16 complex operator M = I (x) G (x) I acting on one 4-bit group,
// with the 2x2 gate G (8 floats: g00r,g00i,g01r,g01i,g10r,g10i,g11r,g11i,
// broadcast-read from LDS) on local bit `wl`. Thread t fills element (t/16,t%16).
__device__ __forceinline__ void buildM(float* mr, float* mi, int wl,
                                       const float* gg, int t) {
    const int row = t >> 4, cm = t & 15;
    const int mask = 15 & ~(1 << wl);
    float re = 0.0f, im = 0.0f;
    if (((row ^ cm) & mask) == 0) {
        const int rb = (row >> wl) & 1;
        const int cb = (cm >> wl) & 1;
        const int e = (rb * 2 + cb) * 2;
        re = gg[e];
        im = gg[e + 1];
    }
    mr[t] = re;
    mi[t] = im;
}

// Apply the 16x16 complex matrix (mr,mi) to the state viewed as S(16 x 256)
// for compile-time group G, via chained V_WMMA_F32_16X16X4_F32.
// Each wave owns 2 disjoint 16-column tiles -> no cross-wave hazard in a gate.
template <int G>
__device__ __forceinline__ void apply16T(float* sr, float* si,
                                         const float* mr, const float* mi,
                                         int wave, int lane) {
    constexpr int SH = G * 4;
    constexpr int LOWMASK = (1 << SH) - 1;
    const int half = lane >> 4;   // 0 for lanes 0-15, 1 for lanes 16-31
    const int n15  = lane & 15;

    // ---- A fragments (gate matrix): identical for both tiles, load once ----
    // A layout: VGPR v, lane l -> M = l%16, K = 4*kc + v + 2*(l/16)
    v2f ar[4], ai[4], ani[4];
#pragma unroll
    for (int kc = 0; kc < 4; ++kc) {
        const int k0 = kc * 4 + 2 * half;
        ar[kc].x  = mr[n15 * 16 + k0];  ar[kc].y  = mr[n15 * 16 + k0 + 1];
        ai[kc].x  = mi[n15 * 16 + k0];  ai[kc].y  = mi[n15 * 16 + k0 + 1];
        ani[kc].x = -ai[kc].x;          ani[kc].y = -ai[kc].y;
    }

#pragma unroll
    for (int tt = 0; tt < 2; ++tt) {
        const int col = (wave * 2 + tt) * 16 + n15;
        const int base = ((col >> SH) << (SH + 4)) | (col & LOWMASK);

        // ---- B fragments (state chunks), all loaded BEFORE in-place store ----
        // B layout: VGPR v, lane l -> K = 4*kc + v + 2*(l/16), N = l%16
        v2f br[4], bi[4];
#pragma unroll
        for (int kc = 0; kc < 4; ++kc) {
            const int k0 = kc * 4 + 2 * half;
            const int i0 = base | ((k0) << SH);
            const int i1 = base | ((k0 + 1) << SH);
            br[kc].x = sr[i0]; br[kc].y = sr[i1];
            bi[kc].x = si[i0]; bi[kc].y = si[i1];
        }

        v8f accR = {0.f,0.f,0.f,0.f,0.f,0.f,0.f,0.f};
        v8f accI = {0.f,0.f,0.f,0.f,0.f,0.f,0.f,0.f};

        // Dr = Mr*Sr - Mi*Si ;  Di = Mr*Si + Mi*Sr
#pragma unroll
        for (int kc = 0; kc < 4; ++kc) {
            accR = __builtin_amdgcn_wmma_f32_16x16x4_f32(false, ar[kc],  false, br[kc], (short)0, accR, false, false);
            accR = __builtin_amdgcn_wmma_f32_16x16x4_f32(false, ani[kc], false, bi[kc], (short)0, accR, false, false);
            accI = __builtin_amdgcn_wmma_f32_16x16x4_f32(false, ar[kc],  false, bi[kc], (short)0, accI, false, false);
            accI = __builtin_amdgcn_wmma_f32_16x16x4_f32(false, ai[kc],  false, br[kc], (short)0, accI, false, false);
        }

        // ---- Store D: VGPR r, lane l -> M = r + 8*(l/16), N = l%16 ----
#pragma unroll
        for (int r = 0; r < 8; ++r) {
            const int i = base | ((r + 8 * half) << SH);
            sr[i] = accR[r];
            si[i] = accI[r];
        }
    }
}

__device__ __forceinline__ void applyDispatch(float* sr, float* si,
                                              const float* mr, const float* mi,
                                              int g, int wave, int lane) {
    if (g == 0)      apply16T<0>(sr, si, mr, mi, wave, lane);
    else if (g == 1) apply16T<1>(sr, si, mr, mi, wave, lane);
    else             apply16T<2>(sr, si, mr, mi, wave, lane);
}

__global__ __launch_bounds__(NTHREADS)
void vqc_kernel(const float* __restrict__ x,      // (B, 12)
                const float* __restrict__ wts,    // (4, 12, 3)
                float* __restrict__ out) {        // (B, 12)
    __shared__ float sr[NSTATE];
    __shared__ float si[NSTATE];
    __shared__ float mr[256];
    __shared__ float mi[256];
    __shared__ float gm[NGATES * 8];   // 2x2 complex gate entries, one row/gate
    __shared__ float wred[NQ * 8];

    const int t    = threadIdx.x;
    const int b    = blockIdx.x;
    const int wave = t >> 5;
    const int lane = t & 31;

    // ---- Precompute all 60 gate 2x2 matrices (one thread per gate) ----
    if (t < NGATES) {
        float e0, e1, e2, e3, e4, e5, e6, e7;
        if (t < NQ) {
            // RX(x[b,t]): [[c, -i s],[-i s, c]]
            const float a = x[b * NQ + t] * 0.5f;
            const float c = cosf(a), s = sinf(a);
            e0 = c;    e1 = 0.f;
            e2 = 0.f;  e3 = -s;
            e4 = 0.f;  e5 = -s;
            e6 = c;    e7 = 0.f;
        } else {
            // Rot(phi,theta,omega), gate index gi = l*12 + q
            const float* wp = wts + (t - NQ) * 3;
            const float phi = wp[0], th = wp[1], om = wp[2];
            const float ct = cosf(th * 0.5f);
            const float sn = sinf(th * 0.5f);
            const float hs = 0.5f * (phi + om);
            const float hd = 0.5f * (phi - om);
            const float chs = cosf(hs), shs = sinf(hs);
            const float chd = cosf(hd), shd = sinf(hd);
            e0 =  chs * ct;  e1 = -shs * ct;   // g00 = e^{-i hs} ct
            e2 = -chd * sn;  e3 = -shd * sn;   // g01 = -e^{+i hd} sn
            e4 =  chd * sn;  e5 = -shd * sn;   // g10 = e^{-i hd} sn
            e6 =  chs * ct;  e7 =  shs * ct;   // g11 = e^{+i hs} ct
        }
        float* gr = gm + t * 8;
        gr[0] = e0; gr[1] = e1; gr[2] = e2; gr[3] = e3;
        gr[4] = e4; gr[5] = e5; gr[6] = e6; gr[7] = e7;
    }

    // ---- |0...0> initial state (stride-256: conflict-free) ----
#pragma unroll
    for (int k = 0; k < 16; ++k) {
        const int i = k * NTHREADS + t;
        sr[i] = 0.0f;
        si[i] = 0.0f;
    }
    if (t == 0) sr[0] = 1.0f;
    __syncthreads();

    // ---------------- AngleEmbedding: RX on wire q (bit 11-q) ----------------
    for (int q = 0; q < NQ; ++q) {
        const int bw = 11 - q;
        buildM(mr, mi, bw & 3, gm + q * 8, t);
        __syncthreads();
        applyDispatch(sr, si, mr, mi, bw >> 2, wave, lane);
        __syncthreads();
    }

    // ---------------- StronglyEntanglingLayers ----------------
    for (int l = 0; l < NLAYERS; ++l) {
        for (int q = 0; q < NQ; ++q) {
            const int bw = 11 - q;
            buildM(mr, mi, bw & 3, gm + (NQ + l * NQ + q) * 8, t);
            __syncthreads();
            applyDispatch(sr, si, mr, mi, bw >> 2, wave, lane);
            __syncthreads();
        }

        // CNOT ring (range r) composed into ONE basis-state permutation:
        // state_after[j] = state_before[ f_0(f_1(...f_11(j))) ],
        // f_w(i) = i ^ (bit_c(i) << t_bit), gates applied in order w = 0..11.
        const int r = l % (NQ - 1) + 1;
        float vr[16], vi[16];
#pragma unroll
        for (int k = 0; k < 16; ++k) {
            const int j = k * NTHREADS + t;
            int src = j;
            for (int qq = NQ - 1; qq >= 0; --qq) {
                const int cbit = 11 - qq;
                const int tbit = 11 - ((qq + r) % NQ);
                src ^= ((src >> cbit) & 1) << tbit;
            }
            vr[k] = sr[src];
            vi[k] = si[src];
        }
        __syncthreads();
#pragma unroll
        for (int k = 0; k < 16; ++k) {
            const int j = k * NTHREADS + t;
            sr[j] = vr[k];
            si[j] = vi[k];
        }
        __syncthreads();
    }

    // ---------------- <Z_q> expectations ----------------
    float part[NQ];
#pragma unroll
    for (int q = 0; q < NQ; ++q) part[q] = 0.0f;
#pragma unroll
    for (int k = 0; k < 16; ++k) {
        const int i = k * NTHREADS + t;
        const float p = sr[i] * sr[i] + si[i] * si[i];
#pragma unroll
        for (int q = 0; q < NQ; ++q) {
            part[q] += ((i >> (11 - q)) & 1) ? -p : p;
        }
    }
#pragma unroll
    for (int q = 0; q < NQ; ++q) {
        float v = part[q];
        for (int off = 16; off >= 1; off >>= 1)
            v += __shfl_xor(v, off, 32);
        if (lane == 0) wred[q * 8 + wave] = v;
    }
    __syncthreads();
    if (t < NQ) {
        float s = 0.0f;
#pragma unroll
        for (int wv = 0; wv < 8; ++wv) s += wred[t * 8 + wv];
        out[b * NQ + t] = s;
    }
}

extern "C" void kernel_launch(void* const* d_in, const int* in_sizes, int n_in,
                              void* d_out, int out_size, void* d_ws, size_t ws_size,
                              hipStream_t stream) {
    const float* x   = (const float*)d_in[0];   // (B, 12) f32
    const float* wts = (const float*)d_in[1];   // (4, 12, 3) f32
    float* out = (float*)d_out;                 // (B, 12) f32
    const int batch = in_sizes[0] / NQ;
    vqc_kernel<<<batch, NTHREADS, 0, stream>>>(x, wts, out);
}